// NFT_MARS_41266045780924
// MI455X (gfx1250) — compile-verified
//
#include <hip/hip_runtime.h>
#include <math.h>

// ---- problem constants (match reference) ----
#define NUM_ITEM  80000
#define NUM_USERS 20000
#define NNODE     100000      // NUM_ITEM + NUM_USERS
#define EDGES     1000000
#define DIMF      64
#define SLOPE     0.01f

typedef float v2f __attribute__((ext_vector_type(2)));
typedef float v8f __attribute__((ext_vector_type(8)));

__device__ __forceinline__ float lrelu(float x) { return x >= 0.f ? x : SLOPE * x; }

// order-preserving float<->uint mapping for atomicMax on signed floats
__device__ __forceinline__ unsigned f2ord(float f) {
  unsigned u = __float_as_uint(f);
  return (u & 0x80000000u) ? ~u : (u | 0x80000000u);
}
__device__ __forceinline__ float ord2f(unsigned u) {
  return (u & 0x80000000u) ? __uint_as_float(u & 0x7FFFFFFFu) : __uint_as_float(~u);
}

// ======================= WMMA GEMM kernels =======================
// V_WMMA_F32_16X16X4_F32 layouts (wave32):
//  A 16x4 : lane L -> row M=L&15 ; VGPR0 = K=(L>>4)*2, VGPR1 = K=(L>>4)*2+1
//  B 4x16 : lane L -> col N=L&15 ; VGPR0 = K=(L>>4)*2, VGPR1 = K=(L>>4)*2+1
//  C 16x16: VGPR r, lane L -> row (L<16 ? r : r+8), col L&15

// xw = X @ W   (X row-major [M,64], W row-major [64,64])
__global__ void k_gemm_nn(const float* __restrict__ X, const float* __restrict__ W,
                          float* __restrict__ C, int M) {
  int wave = blockIdx.x * (blockDim.x >> 5) + (threadIdx.x >> 5);
  int lane = threadIdx.x & 31;
  int rb = wave >> 2, ct = wave & 3;
  int row0 = rb * 16;
  if (row0 >= M) return;
  int ml = lane & 15, half = lane >> 4;
  int m = row0 + ml;
  int n = ct * 16 + ml;
  const float* arow = X + (size_t)m * DIMF;
  v8f c = {};
  #pragma unroll
  for (int k = 0; k < DIMF; k += 4) {
    int kk = k + half * 2;
    v2f a, b;
    a.x = arow[kk];           a.y = arow[kk + 1];
    b.x = W[kk * DIMF + n];   b.y = W[(kk + 1) * DIMF + n];
    c = __builtin_amdgcn_wmma_f32_16x16x4_f32(false, a, false, b, (short)0, c, false, false);
  }
  #pragma unroll
  for (int r = 0; r < 8; ++r) {
    int rr = row0 + r + half * 8;
    C[(size_t)rr * DIMF + n] = c[r];
  }
}

// U = user_features @ user_mlp_w^T  (A [M,128], W [64,128]); raw product out.
__global__ void k_gemm_user(const float* __restrict__ A, const float* __restrict__ W,
                            float* __restrict__ C, int M) {
  int wave = blockIdx.x * (blockDim.x >> 5) + (threadIdx.x >> 5);
  int lane = threadIdx.x & 31;
  int rb = wave >> 2, ct = wave & 3;
  int row0 = rb * 16;
  if (row0 >= M) return;
  int ml = lane & 15, half = lane >> 4;
  int m = row0 + ml;
  int f = ct * 16 + ml;
  const float* arow = A + (size_t)m * 128;
  const float* wrow = W + (size_t)f * 128;
  v8f c = {};
  #pragma unroll
  for (int k = 0; k < 128; k += 4) {
    int kk = k + half * 2;
    v2f a, b;
    a.x = arow[kk]; a.y = arow[kk + 1];
    b.x = wrow[kk]; b.y = wrow[kk + 1];
    c = __builtin_amdgcn_wmma_f32_16x16x4_f32(false, a, false, b, (short)0, c, false, false);
  }
  #pragma unroll
  for (int r = 0; r < 8; ++r) {
    int rr = row0 + r + half * 8;
    C[(size_t)rr * DIMF + f] = c[r];
  }
}

// Fused tail: x_out = lrelu( lrelu(h)@gW^T + gB + lrelu(x@linW^T + linB) + id )
// Two interleaved WMMA accumulator chains; epilogue writes scratch and d_out slice.
__global__ void k_tail(const float* __restrict__ xin, const float* __restrict__ hin,
                       const float* __restrict__ gW, const float* __restrict__ gB,
                       const float* __restrict__ linW, const float* __restrict__ linB,
                       const float* __restrict__ idw, float* __restrict__ xout,
                       float* __restrict__ dout, int colofs) {
  int wave = blockIdx.x * (blockDim.x >> 5) + (threadIdx.x >> 5);
  int lane = threadIdx.x & 31;
  int rb = wave >> 2, ct = wave & 3;
  int row0 = rb * 16;
  if (row0 >= NNODE) return;
  int ml = lane & 15, half = lane >> 4;
  int m = row0 + ml;
  int f = ct * 16 + ml;
  const float* hrow = hin + (size_t)m * DIMF;
  const float* xrow = xin + (size_t)m * DIMF;
  const float* grow = gW + (size_t)f * DIMF;
  const float* lrow = linW + (size_t)f * DIMF;
  v8f c1 = {}, c2 = {};
  #pragma unroll
  for (int k = 0; k < DIMF; k += 4) {
    int kk = k + half * 2;
    v2f a1, a2, b1, b2;
    a1.x = lrelu(hrow[kk]); a1.y = lrelu(hrow[kk + 1]);
    a2.x = xrow[kk];        a2.y = xrow[kk + 1];
    b1.x = grow[kk];        b1.y = grow[kk + 1];
    b2.x = lrow[kk];        b2.y = lrow[kk + 1];
    c1 = __builtin_amdgcn_wmma_f32_16x16x4_f32(false, a1, false, b1, (short)0, c1, false, false);
    c2 = __builtin_amdgcn_wmma_f32_16x16x4_f32(false, a2, false, b2, (short)0, c2, false, false);
  }
  float gb = gB[f], lb = linB[f];
  #pragma unroll
  for (int r = 0; r < 8; ++r) {
    int rr = row0 + r + half * 8;
    float t = lrelu(c2[r] + lb);
    float v = lrelu(c1[r] + gb + t + idw[(size_t)rr * DIMF + f]);
    xout[(size_t)rr * DIMF + f] = v;
    dout[(size_t)rr * 128 + colofs + f] = v;
  }
}

// ======================= node prep / init =======================
// Build x = normalize(concat(features, tanh(U + b))) ; one wave per row.
__global__ void k_build_x(const float* __restrict__ feat, const float* __restrict__ U,
                          const float* __restrict__ ub, float* __restrict__ x) {
  int row = blockIdx.x * (blockDim.x >> 5) + (threadIdx.x >> 5);
  int lane = threadIdx.x & 31;
  if (row >= NNODE) return;
  int col = lane * 2;
  float v0, v1;
  if (row < NUM_ITEM) {
    v0 = feat[(size_t)row * DIMF + col];
    v1 = feat[(size_t)row * DIMF + col + 1];
  } else {
    int u = row - NUM_ITEM;
    v0 = tanhf(U[(size_t)u * DIMF + col] + ub[col]);
    v1 = tanhf(U[(size_t)u * DIMF + col + 1] + ub[col + 1]);
  }
  float s = v0 * v0 + v1 * v1;
  #pragma unroll
  for (int o = 16; o >= 1; o >>= 1) s += __shfl_xor(s, o, 32);
  float scale = 1.0f / fmaxf(sqrtf(s), 1e-12f);
  x[(size_t)row * DIMF + col] = v0 * scale;
  x[(size_t)row * DIMF + col + 1] = v1 * scale;
}

__global__ void k_zero(float* __restrict__ p, int n) {
  int i = blockIdx.x * blockDim.x + threadIdx.x;
  if (i < n) p[i] = 0.f;
}

__global__ void k_deg(const int* __restrict__ ei, float* __restrict__ deg) {
  int e = blockIdx.x * blockDim.x + threadIdx.x;
  if (e < EDGES) atomicAdd(&deg[ei[e]], 1.0f);
}

__global__ void k_init_hop(float* __restrict__ h, unsigned* __restrict__ maxu,
                           float* __restrict__ sumz) {
  int i = blockIdx.x * blockDim.x + threadIdx.x;
  if (i < NNODE * DIMF) h[i] = 0.f;
  if (i < NNODE) { maxu[i] = 0u; sumz[i] = 0.f; }
}

// ======================= edge passes (L2-resident gathers) =======================
// Pass 1: logits + segment-max.  16 lanes per edge, float4 gathers.
__global__ void k_edge_logit(const int* __restrict__ ei, const float* __restrict__ xw,
                             const float* __restrict__ deg, float* __restrict__ logits,
                             unsigned* __restrict__ maxu) {
  int t = blockIdx.x * blockDim.x + threadIdx.x;
  int e = t >> 4, l = t & 15;
  if (e >= EDGES) return;
  int s = ei[e], d = ei[EDGES + e];
  const float4* ps = (const float4*)(xw + (size_t)s * DIMF);
  const float4* pd = (const float4*)(xw + (size_t)d * DIMF);
  float4 a = ps[l], b = pd[l];
  float p = a.x * b.x + a.y * b.y + a.z * b.z + a.w * b.w;
  #pragma unroll
  for (int o = 8; o >= 1; o >>= 1) p += __shfl_xor(p, o, 32);
  if (l == 0) {
    float dinv = rsqrtf(deg[s]);
    float gate = 1.0f / (1.0f + expf(-dinv * p));
    float logit = p * gate;
    logits[e] = logit;
    atomicMax(&maxu[d], f2ord(logit));
  }
}

// Pass 2: e = exp(logit - m[dst]); segment-sum.
__global__ void k_edge_exp(const int* __restrict__ ei, float* __restrict__ logits,
                           const unsigned* __restrict__ maxu, float* __restrict__ sumz) {
  int e = blockIdx.x * blockDim.x + threadIdx.x;
  if (e >= EDGES) return;
  int d = ei[EDGES + e];
  float m = ord2f(maxu[d]);
  float ex = expf(logits[e] - m);
  logits[e] = ex;
  atomicAdd(&sumz[d], ex);
}

// Pass 3: scatter-add  h[dst] += xw[src] * attn.
__global__ void k_edge_scatter(const int* __restrict__ ei, const float* __restrict__ xw,
                               const float* __restrict__ logits, const float* __restrict__ sumz,
                               float* __restrict__ h) {
  int t = blockIdx.x * blockDim.x + threadIdx.x;
  int e = t >> 4, l = t & 15;
  if (e >= EDGES) return;
  int s = ei[e], d = ei[EDGES + e];
  float attn = logits[e] / (sumz[d] + 1e-16f);
  const float4* ps = (const float4*)(xw + (size_t)s * DIMF);
  float4 a = ps[l];
  float* hd = h + (size_t)d * DIMF + l * 4;
  atomicAdd(hd + 0, a.x * attn);
  atomicAdd(hd + 1, a.y * attn);
  atomicAdd(hd + 2, a.z * attn);
  atomicAdd(hd + 3, a.w * attn);
}

// ======================= host side =======================
static void run_hop(const float* xin, const float* gatW,
                    const float* linW, const float* linB,
                    const float* gW, const float* gB,
                    const float* idw, const int* ei,
                    float* xw, float* hbuf, float* logits, float* deg,
                    unsigned* maxu, float* sumz,
                    float* xout, float* dout, int colofs, hipStream_t stream) {
  // xw = xin @ gatW
  k_gemm_nn<<<(NNODE / 16 * 4 + 3) / 4, 128, 0, stream>>>(xin, gatW, xw, NNODE);
  // h = 0, maxu = 0, sumz = 0
  k_init_hop<<<(NNODE * DIMF + 255) / 256, 256, 0, stream>>>(hbuf, maxu, sumz);
  // edge passes
  k_edge_logit<<<(EDGES * 16 + 255) / 256, 256, 0, stream>>>(ei, xw, deg, logits, maxu);
  k_edge_exp<<<(EDGES + 255) / 256, 256, 0, stream>>>(ei, logits, maxu, sumz);
  k_edge_scatter<<<(EDGES * 16 + 255) / 256, 256, 0, stream>>>(ei, xw, logits, sumz, hbuf);
  // fused GEMM tail + epilogue (+ direct write into output slice)
  k_tail<<<(NNODE / 16 * 4 + 3) / 4, 128, 0, stream>>>(xin, hbuf, gW, gB, linW, linB,
                                                       idw, xout, dout, colofs);
}

extern "C" void kernel_launch(void* const* d_in, const int* in_sizes, int n_in,
                              void* d_out, int out_size, void* d_ws, size_t ws_size,
                              hipStream_t stream) {
  const float* features = (const float*)d_in[0];
  const float* user_feats = (const float*)d_in[1];
  const float* idw   = (const float*)d_in[2];
  const float* umw   = (const float*)d_in[3];
  const float* umb   = (const float*)d_in[4];
  const float* gat1  = (const float*)d_in[5];
  const float* lin1w = (const float*)d_in[6];
  const float* lin1b = (const float*)d_in[7];
  const float* g1w   = (const float*)d_in[8];
  const float* g1b   = (const float*)d_in[9];
  const float* gat2  = (const float*)d_in[10];
  const float* lin2w = (const float*)d_in[11];
  const float* lin2b = (const float*)d_in[12];
  const float* g2w   = (const float*)d_in[13];
  const float* g2b   = (const float*)d_in[14];
  const int*   ei    = (const int*)d_in[15];
  float* out = (float*)d_out;

  // workspace layout (floats)
  float* ws = (float*)d_ws;
  const size_t NF = (size_t)NNODE * DIMF;
  float* xbuf   = ws;                    // x (hop1 in), reused as x2 sink
  float* x1buf  = xbuf + NF;             // x1
  float* xw     = x1buf + NF;            // per-hop x@W (also user-MLP scratch)
  float* hbuf   = xw + NF;               // aggregated messages
  float* logits = hbuf + NF;             // per-edge logits / exp
  float* deg    = logits + EDGES;        // src degree
  unsigned* maxu = (unsigned*)(deg + NNODE);
  float* sumz   = (float*)(maxu + NNODE);

  // --- phase 0: user MLP (WMMA), concat+tanh+normalize, degrees ---
  k_gemm_user<<<(NUM_USERS / 16 * 4 + 3) / 4, 128, 0, stream>>>(user_feats, umw, xw, NUM_USERS);
  k_build_x<<<(NNODE + 7) / 8, 256, 0, stream>>>(features, xw, umb, xbuf);
  k_zero<<<(NNODE + 255) / 256, 256, 0, stream>>>(deg, NNODE);
  k_deg<<<(EDGES + 255) / 256, 256, 0, stream>>>(ei, deg);

  // --- hop 1: writes x1buf and out[:, 0:64] ---
  run_hop(xbuf, gat1, lin1w, lin1b, g1w, g1b, idw, ei,
          xw, hbuf, logits, deg, maxu, sumz, x1buf, out, 0, stream);
  // --- hop 2: input x1buf, writes out[:, 64:128] ---
  run_hop(x1buf, gat2, lin2w, lin2b, g2w, g2b, idw, ei,
          xw, hbuf, logits, deg, maxu, sumz, xbuf, out, 64, stream);
}